// soft_IHT_layer_23270132810499
// MI455X (gfx1250) — compile-verified
//
#include <hip/hip_runtime.h>
#include <hip/hip_bf16.h>

// Problem constants (match reference)
#define MM   512
#define NN   1024
#define BB   64
#define SSP  50
#define ETA  0.1f

typedef __attribute__((ext_vector_type(2))) float v2f;
typedef __attribute__((ext_vector_type(8))) float v8f;
typedef int v4i __attribute__((vector_size(16)));   // builtin's b128 payload type

#define AS1 __attribute__((address_space(1)))
#define AS3 __attribute__((address_space(3)))

// Wait for outstanding async global->LDS transfers issued by this wave, then
// workgroup barrier. Safe (counter is 0) when the sync staging path was used.
__device__ __forceinline__ void wait_async_then_barrier() {
#if __has_builtin(__builtin_amdgcn_s_wait_asynccnt)
    __builtin_amdgcn_s_wait_asynccnt(0);
#else
    asm volatile("s_wait_asynccnt 0x0" ::: "memory");
#endif
    __syncthreads();
}

// ---------------------------------------------------------------------------
// Kernel 1: R = Y^T - A @ X      A:(512,1024) X:(1024,64) Y:(64,512) R:(512,64)
// One block per 16-row strip of A (32 blocks). 4 waves/block, wave w owns the
// 16-col tile n0 = 16w. A strip staged in LDS (padded stride 516 -> 32-lane
// conflict-free fragment reads), processed in two K-chunks of 512. Staging
// uses CDNA5 async global->LDS (ASYNCcnt) when the builtin is available.
// ---------------------------------------------------------------------------
__global__ __launch_bounds__(128) void gemm1_AX(const float* __restrict__ A,
                                                const float* __restrict__ X,
                                                const float* __restrict__ Y,
                                                float* __restrict__ R) {
    __shared__ float sA[16 * 516];             // 33 KB, stride 516 (516%64==4)
    const int t    = threadIdx.x;
    const int wave = t >> 5;
    const int lane = t & 31;
    const int lr   = lane & 15;                // row within A-frag / col within C
    const int hi   = lane >> 4;
    const int ko   = hi * 2;                   // K sub-offset per ISA A layout
    const int m0   = blockIdx.x * 16;
    const int n0   = wave * 16;

    v8f acc = {};
    for (int kk = 0; kk < NN; kk += 512) {
        __syncthreads();                       // previous chunk fully consumed
#if __has_builtin(__builtin_amdgcn_global_load_async_to_lds_b128)
        // Async DMA of the 16x512 strip: 16 B per lane per issue, no VGPR
        // round-trip; LDS rows padded to 516 floats (2064 B, 16 B aligned).
        for (int q = t; q < 16 * 128; q += 128) {
            int row  = q >> 7;                 // 128 float4 per row
            int col4 = q & 127;
            AS1 v4i* gsrc = (AS1 v4i*)(A + (m0 + row) * NN + kk + col4 * 4);
            AS3 v4i* ldst = (AS3 v4i*)(&sA[row * 516 + col4 * 4]);
            __builtin_amdgcn_global_load_async_to_lds_b128(gsrc, ldst, 0, 0);
        }
#else
        // Fallback: synchronous coalesced staging
        for (int idx = t; idx < 16 * 512; idx += 128) {
            int row = idx >> 9;
            int col = idx & 511;
            sA[row * 516 + col] = A[(m0 + row) * NN + kk + col];
        }
#endif
        wait_async_then_barrier();

        #pragma unroll 4
        for (int k0 = 0; k0 < 512; k0 += 4) {
            v2f a;                              // A-frag: v0=K{0,2}, v1=K{1,3}
            const float* pa = &sA[lr * 516 + k0 + ko];
            a.x = pa[0];
            a.y = pa[1];
            v2f b;                              // B-frag from X (K x 16 tile)
            const int kr = kk + k0 + ko;
            b.x = X[kr * BB + n0 + lr];
            b.y = X[(kr + 1) * BB + n0 + lr];
            acc = __builtin_amdgcn_wmma_f32_16x16x4_f32(
                false, a, false, b, (short)0, acc, false, false);
        }
    }
    // C/D layout: VGPR r -> rows m0+r (lanes 0-15) and m0+r+8 (lanes 16-31)
    #pragma unroll
    for (int r = 0; r < 8; ++r) {
        int row = m0 + r + hi * 8;
        int col = n0 + lr;
        R[row * BB + col] = Y[col * MM + row] - acc[r];
    }
}

// ---------------------------------------------------------------------------
// Kernel 2: H1 = X + eta * (A^T @ R)    A^T:(1024,512) R:(512,64) H1:(1024,64)
// A^T fragment reads are 16 contiguous floats per half-wave -> coalesced,
// no LDS staging needed. 64 blocks x 4 waves cover the 64x4 tile grid.
// global_prefetch pulls the strided A rows ~64 K-steps ahead.
// ---------------------------------------------------------------------------
__global__ __launch_bounds__(128) void gemm2_AtR(const float* __restrict__ A,
                                                 const float* __restrict__ R,
                                                 const float* __restrict__ X,
                                                 float* __restrict__ H1) {
    const int t    = threadIdx.x;
    const int wave = t >> 5;
    const int lane = t & 31;
    const int lr   = lane & 15;
    const int hi   = lane >> 4;
    const int ko   = hi * 2;
    const int n0   = blockIdx.x * 16;          // row tile in N-dim
    const int b0   = wave * 16;                // col tile in B-dim

    v8f acc = {};
    #pragma unroll 4
    for (int k0 = 0; k0 < MM; k0 += 4) {
        const int kr = k0 + ko;
        if (((k0 & 15) == 0) && (k0 + 64 < MM))
            __builtin_prefetch(&A[(k0 + 64 + ko) * NN + n0 + lr], 0, 1);
        v2f a;                                  // A^T[n0+lr][kr..kr+1]
        a.x = A[kr * NN + n0 + lr];
        a.y = A[(kr + 1) * NN + n0 + lr];
        v2f b;
        b.x = R[kr * BB + b0 + lr];
        b.y = R[(kr + 1) * BB + b0 + lr];
        acc = __builtin_amdgcn_wmma_f32_16x16x4_f32(
            false, a, false, b, (short)0, acc, false, false);
    }
    #pragma unroll
    for (int r = 0; r < 8; ++r) {
        int row = n0 + r + hi * 8;
        int col = b0 + lr;
        int idx = row * BB + col;
        H1[idx] = X[idx] + ETA * acc[r];
    }
}

// ---------------------------------------------------------------------------
// Kernel 3: per-batch NeuralSort soft top-S mask; out = mask^T * H1.
// One block (8 wave32s) per batch sample. z in LDS (broadcast-read in the
// O(N^2) pairwise loop), c_i in registers, softmax reductions via wave32
// shuffles + 8-slot LDS cross-wave combine. TAU == 1 folds away.
// ---------------------------------------------------------------------------
__global__ __launch_bounds__(256) void mask_kernel(const float* __restrict__ H1,
                                                   const float* __restrict__ W,
                                                   float* __restrict__ out) {
    __shared__ float zs[NN];
    __shared__ float red[8];
    const int b    = blockIdx.x;
    const int t    = threadIdx.x;
    const int wave = t >> 5;
    const int lane = t & 31;

    float h[4], z[4], c[4], msk[4];
    #pragma unroll
    for (int q = 0; q < 4; ++q) {
        int i = t + 256 * q;
        h[q] = H1[i * BB + b];
        z[q] = fabsf(W[i] * h[q]);
        zs[i] = z[q];
        msk[q] = 0.0f;
    }
    __syncthreads();

    // c_i = sum_j |z_i - z_j|  (LDS broadcast reads, no bank conflicts)
    #pragma unroll
    for (int q = 0; q < 4; ++q) {
        float zi = z[q];
        float accv = 0.0f;
        #pragma unroll 4
        for (int j = 0; j < NN; ++j) accv += fabsf(zi - zs[j]);
        c[q] = accv;
    }

    for (int s = 0; s < SSP; ++s) {
        const float ks = (float)(NN + 1 - 2 * (s + 1));
        float l[4];
        float mx = -3.0e38f;
        #pragma unroll
        for (int q = 0; q < 4; ++q) {
            l[q] = ks * z[q] - c[q];
            mx = fmaxf(mx, l[q]);
        }
        #pragma unroll
        for (int off = 16; off > 0; off >>= 1)
            mx = fmaxf(mx, __shfl_xor(mx, off, 32));
        if (lane == 0) red[wave] = mx;
        __syncthreads();
        float gmx = red[0];
        #pragma unroll
        for (int w = 1; w < 8; ++w) gmx = fmaxf(gmx, red[w]);
        __syncthreads();

        float e[4], sm = 0.0f;
        #pragma unroll
        for (int q = 0; q < 4; ++q) {
            e[q] = __expf(l[q] - gmx);
            sm += e[q];
        }
        #pragma unroll
        for (int off = 16; off > 0; off >>= 1)
            sm += __shfl_xor(sm, off, 32);
        if (lane == 0) red[wave] = sm;
        __syncthreads();
        float gs = 0.0f;
        #pragma unroll
        for (int w = 0; w < 8; ++w) gs += red[w];
        __syncthreads();

        const float inv = 1.0f / gs;
        #pragma unroll
        for (int q = 0; q < 4; ++q) msk[q] += e[q] * inv;
    }

    #pragma unroll
    for (int q = 0; q < 4; ++q) {
        int i = t + 256 * q;
        out[i * BB + b] = msk[q] * h[q];
    }
}

// ---------------------------------------------------------------------------
extern "C" void kernel_launch(void* const* d_in, const int* in_sizes, int n_in,
                              void* d_out, int out_size, void* d_ws, size_t ws_size,
                              hipStream_t stream) {
    (void)in_sizes; (void)n_in; (void)out_size; (void)ws_size;
    const float* X = (const float*)d_in[0];   // (N,B)
    const float* Y = (const float*)d_in[1];   // (B,M)
    const float* A = (const float*)d_in[2];   // (M,N)
    const float* W = (const float*)d_in[3];   // (N,)
    float* out = (float*)d_out;               // (N,B)

    float* R  = (float*)d_ws;                 // (M,B)  = 512*64 floats
    float* H1 = R + MM * BB;                  // (N,B)  = 1024*64 floats

    gemm1_AX <<<MM / 16, 128, 0, stream>>>(A, X, Y, R);
    gemm2_AtR<<<NN / 16, 128, 0, stream>>>(A, R, X, H1);
    mask_kernel<<<BB, 256, 0, stream>>>(H1, W, out);
}